// Attention_Pooling_GNN_25520695673360
// MI455X (gfx1250) — compile-verified
//
#include <hip/hip_runtime.h>
#include <hip/hip_bf16.h>

#define PATCHF     202.0f
#define NP         6
#define NCELL      36      // 6*6
#define BATCH      32
#define C          128
#define TWO_C      256
#define HDIM       128
#define ROWS       (BATCH * NCELL)   // 1152

typedef __attribute__((ext_vector_type(2))) float v2f;
typedef __attribute__((ext_vector_type(8))) float v8f;

// ---------------------------------------------------------------------------
// Stage 0: zero the pooled accumulator P[1152][256] in workspace.
// ---------------------------------------------------------------------------
__global__ void zero_f32(float* __restrict__ p, int n) {
    int i = blockIdx.x * blockDim.x + threadIdx.x;
    if (i < n) p[i] = 0.0f;
}

// ---------------------------------------------------------------------------
// Stage 1: segment-sum scatter pool. One wave (32 lanes) per point; each lane
// handles 4 contiguous channels (float4 load -> 4 fp32 atomics into L2-resident
// accumulator). row = batch*36 + (px*6+py); column offset selects graph half.
// ---------------------------------------------------------------------------
__global__ void pool_scatter(const float* __restrict__ x,
                             const float* __restrict__ pos,
                             const int*   __restrict__ batch,
                             float* __restrict__ P,
                             int N, int colOff) {
    int t     = blockIdx.x * blockDim.x + threadIdx.x;
    int point = t >> 5;
    int lane  = t & 31;
    if (point >= N) return;

    float fx = pos[(size_t)point * 2 + 0];
    float fy = pos[(size_t)point * 2 + 1];
    int px = (int)floorf(fx / PATCHF);
    int py = (int)floorf(fy / PATCHF);
    px = px < 0 ? 0 : (px > NP - 1 ? NP - 1 : px);
    py = py < 0 ? 0 : (py > NP - 1 ? NP - 1 : py);
    int bt  = batch[point];
    int row = bt * NCELL + (px * NP + py);

    const float4 v = *(const float4*)(x + (size_t)point * C + lane * 4);
    float* dst = P + (size_t)row * TWO_C + colOff + lane * 4;
    __hip_atomic_fetch_add(dst + 0, v.x, __ATOMIC_RELAXED, __HIP_MEMORY_SCOPE_AGENT);
    __hip_atomic_fetch_add(dst + 1, v.y, __ATOMIC_RELAXED, __HIP_MEMORY_SCOPE_AGENT);
    __hip_atomic_fetch_add(dst + 2, v.z, __ATOMIC_RELAXED, __HIP_MEMORY_SCOPE_AGENT);
    __hip_atomic_fetch_add(dst + 3, v.w, __ATOMIC_RELAXED, __HIP_MEMORY_SCOPE_AGENT);
}

// ---------------------------------------------------------------------------
// Stage 2: E[1152][128] = P[1152][256] @ W_emb[256][128] + b_emb, via
// V_WMMA_F32_16X16X4_F32. One wave per 16x16 output tile, 64 K-steps.
//
// A 16x4 fp32 layout (per ISA 7.12.2): lane = hi*16+lo; VGPR0 = K=2*hi,
// VGPR1 = K=2*hi+1, row M = lo  -> contiguous float2 from P row.
// B 4x16 fp32: VGPR0/1 hold K=2*hi / K=2*hi+1, col N = lo -> strided loads.
// D: VGPR i -> row (i + 8*hi), col lo.
// ---------------------------------------------------------------------------
__global__ void embed_wmma(const float* __restrict__ P,
                           const float* __restrict__ Wemb,
                           const float* __restrict__ bemb,
                           float* __restrict__ E) {
    const int mt   = blockIdx.x;        // 0..71
    const int nt   = blockIdx.y;        // 0..7
    const int lane = threadIdx.x;       // 0..31, full wave
    const int hi   = lane >> 4;         // 0 or 1
    const int lo   = lane & 15;

    const int m = mt * 16 + lo;
    const int n = nt * 16 + lo;

    const float* arow = P + (size_t)m * TWO_C + 2 * hi;          // K=2*hi, 2*hi+1
    const float* bcol = Wemb + (size_t)(2 * hi) * HDIM + n;      // rows 2*hi(+1)

    v8f acc = {0.f, 0.f, 0.f, 0.f, 0.f, 0.f, 0.f, 0.f};

    for (int kk = 0; kk < TWO_C; kk += 4) {
        v2f a = *(const v2f*)(arow + kk);          // P[m][kk+2hi], P[m][kk+2hi+1]
        v2f b;
        b.x = bcol[(size_t)kk * HDIM];             // Wemb[kk+2hi][n]
        b.y = bcol[(size_t)(kk + 1) * HDIM];       // Wemb[kk+2hi+1][n]
        acc = __builtin_amdgcn_wmma_f32_16x16x4_f32(
            /*neg_a=*/false, a, /*neg_b=*/false, b,
            /*c_mod=*/(short)0, acc, /*reuse_a=*/false, /*reuse_b=*/false);
    }

    const float bias = bemb[n];
#pragma unroll
    for (int i = 0; i < 8; ++i) {
        int mrow = mt * 16 + i + 8 * hi;
        E[(size_t)mrow * HDIM + n] = acc[i] + bias;
    }
}

// ---------------------------------------------------------------------------
// Stage 3: per-batch attention: scores -> softmax over 36 cells -> weighted
// sum of embeddings -> 128x2 FC. One 128-thread block per batch.
// ---------------------------------------------------------------------------
__global__ void attn_fc(const float* __restrict__ E,
                        const float* __restrict__ Watt,
                        const float* __restrict__ batt,
                        const float* __restrict__ Wfc,
                        const float* __restrict__ bfc,
                        float* __restrict__ out) {
    const int b = blockIdx.x;
    const int t = threadIdx.x;          // 0..127

    __shared__ float sc[NCELL];
    __shared__ float att[HDIM];

    if (t < NCELL) {
        const float* row = E + (size_t)(b * NCELL + t) * HDIM;
        float s = batt[0];
        for (int h = 0; h < HDIM; ++h) s += row[h] * Watt[h];
        sc[t] = s;
    }
    __syncthreads();

    if (t == 0) {
        float mx = sc[0];
        for (int i = 1; i < NCELL; ++i) mx = fmaxf(mx, sc[i]);
        float sum = 0.0f;
        for (int i = 0; i < NCELL; ++i) { float e = expf(sc[i] - mx); sc[i] = e; sum += e; }
        float inv = 1.0f / sum;
        for (int i = 0; i < NCELL; ++i) sc[i] *= inv;
    }
    __syncthreads();

    {
        float a = 0.0f;
        for (int cell = 0; cell < NCELL; ++cell)
            a += sc[cell] * E[(size_t)(b * NCELL + cell) * HDIM + t];
        att[t] = a;
    }
    __syncthreads();

    if (t < 2) {
        float o = bfc[t];
        for (int h = 0; h < HDIM; ++h) o += att[h] * Wfc[h * 2 + t];
        out[b * 2 + t] = o;
    }
}

// ---------------------------------------------------------------------------
extern "C" void kernel_launch(void* const* d_in, const int* in_sizes, int n_in,
                              void* d_out, int out_size, void* d_ws, size_t ws_size,
                              hipStream_t stream) {
    const float* x1   = (const float*)d_in[0];
    const float* x2   = (const float*)d_in[1];
    const float* pos1 = (const float*)d_in[2];
    const float* pos2 = (const float*)d_in[3];
    const int*   b1   = (const int*)d_in[4];
    const int*   b2   = (const int*)d_in[5];
    const float* Wemb = (const float*)d_in[6];
    const float* bemb = (const float*)d_in[7];
    const float* Watt = (const float*)d_in[8];
    const float* batt = (const float*)d_in[9];
    const float* Wfc  = (const float*)d_in[10];
    const float* bfc  = (const float*)d_in[11];

    const int N = in_sizes[0] / C;

    float* P = (float*)d_ws;                       // [1152][256]
    float* E = P + (size_t)ROWS * TWO_C;           // [1152][128]

    const int nP = ROWS * TWO_C;
    zero_f32<<<(nP + 255) / 256, 256, 0, stream>>>(P, nP);

    const int threads = N * 32;                    // one wave per point
    const int blocks  = (threads + 255) / 256;
    pool_scatter<<<blocks, 256, 0, stream>>>(x1, pos1, b1, P, N, 0);
    pool_scatter<<<blocks, 256, 0, stream>>>(x2, pos2, b2, P, N, C);

    dim3 gg(ROWS / 16, HDIM / 16);                 // 72 x 8 tiles
    embed_wmma<<<gg, 32, 0, stream>>>(P, Wemb, bemb, E);

    attn_fc<<<BATCH, HDIM, 0, stream>>>(E, Watt, batt, Wfc, bfc, (float*)d_out);
}